// Attention_UNet_MaskBox_ROIMaskRCNN_22952305230527
// MI455X (gfx1250) — compile-verified
//
#include <hip/hip_runtime.h>
#include <stdint.h>

// Mask R-CNN paste_mask_in_image for MI455X (gfx1250).
// Memory-bound: 128 MB output stream -> ~5.6 us floor at 23.3 TB/s.
// CDNA5 paths: global_load_async_to_lds_b128 (ASYNCcnt) staging of masks,
// s_wait_asynccnt, LDS bilinear gathers, non-temporal b128 output stores.

#define BIMG   32
#define NBOX   16
#define MDIM   28
#define MPAD   30
#define HIMG   1024
#define WIMG   1024
#define ROWS   32          // rows of the image per block (amortize mask staging)
#define NTHR   512         // threads per block (16 waves)
#define CHUNKS ((ROWS * WIMG) / (4 * NTHR))   // float4 chunks per thread = 16

typedef float v4f __attribute__((ext_vector_type(4)));

__global__ __launch_bounds__(NTHR)
void paste_masks_kernel(const float* __restrict__ masks,   // [B,N,1,28,28]
                        const float* __restrict__ rects,   // [B,N,4] (x0,y0,x1,y1)
                        float* __restrict__ out)           // [B,1,H,W]
{
  __shared__ float    sraw[NBOX * MDIM * MDIM];   // 50176 B: async-load target
  __shared__ float    smp [NBOX * MPAD * MPAD];   // 57600 B: padded, (m+1)/2 masks
  __shared__ int      sb0[NBOX], sb1[NBOX], sb2[NBOX], sb3[NBOX];
  __shared__ float    sfx[NBOX], sfy[NBOX];
  __shared__ unsigned srowmask[ROWS];             // bit n: box n covers this row
  __shared__ int      srx0[NBOX * ROWS], srx1[NBOX * ROWS];
  __shared__ float    srtx[NBOX * ROWS];

  const int tid  = threadIdx.x;
  const int bid  = blockIdx.x;
  const int b    = bid / (HIMG / ROWS);
  const int rg   = bid % (HIMG / ROWS);
  const int row0 = rg * ROWS;

  // ---- Phase 0: async-stage this image's 16 masks (50 KB) into LDS --------
  {
    const float* src = masks + (size_t)b * (NBOX * MDIM * MDIM);
    unsigned lds_base = (unsigned)(uintptr_t)(&sraw[0]);
    for (int i = tid; i < (NBOX * MDIM * MDIM) / 4; i += NTHR) {
      unsigned ldsa = lds_base + (unsigned)i * 16u;
      unsigned goff = (unsigned)i * 16u;
      asm volatile("global_load_async_to_lds_b128 %0, %1, %2"
                   :: "v"(ldsa), "v"(goff), "s"(src) : "memory");
    }
  }

  // ---- Phase 1: expanded + truncated box params ---------------------------
  if (tid < NBOX) {
    const float* r = rects + ((size_t)b * NBOX + tid) * 4;
    float x0 = r[0], y0 = r[1], x1 = r[2], y1 = r[3];
    const float scale = (float)MPAD / (float)MDIM;         // (M+2)/M
    float w_half = (x1 - x0) * (0.5f * scale);
    float h_half = (y1 - y0) * (0.5f * scale);
    float xc = (x1 + x0) * 0.5f;
    float yc = (y1 + y0) * 0.5f;
    int b0 = (int)truncf(xc - w_half);   // row-min (reference uses x along H)
    int b1 = (int)truncf(yc - h_half);   // col-min
    int b2 = (int)truncf(xc + w_half);   // row-max
    int b3 = (int)truncf(yc + h_half);   // col-max
    float w = (float)max(b2 - b0 + 1, 1);
    float h = (float)max(b3 - b1 + 1, 1);
    sb0[tid] = b0; sb1[tid] = b1; sb2[tid] = b2; sb3[tid] = b3;
    sfx[tid] = (float)MPAD / w;
    sfy[tid] = (float)MPAD / h;
  }
  __syncthreads();

  // ---- Phase 2: per-row box bitmask + per-(box,row) x-interp table --------
  if (tid < ROWS) {
    int i = row0 + tid;
    unsigned m = 0;
    #pragma unroll
    for (int n = 0; n < NBOX; ++n)
      if (i >= sb0[n] && i <= sb2[n]) m |= (1u << n);
    srowmask[tid] = m;
  }
  if (tid < NBOX * ROWS) {
    int n = tid / ROWS, r = tid % ROWS;
    int i = row0 + r;
    float s = ((float)(i - sb0[n]) + 0.5f) * sfx[n] - 0.5f;
    s = fminf(fmaxf(s, 0.0f), (float)(MPAD - 1));
    float f = floorf(s);
    int x0 = (int)f;
    srx0[tid] = x0;
    srx1[tid] = min(x0 + 1, MPAD - 1);
    srtx[tid] = s - f;
  }

  asm volatile("s_wait_asynccnt 0" ::: "memory");
  __syncthreads();

  // ---- Phase 3: padded, [0,1]-mapped masks in LDS -------------------------
  for (int idx = tid; idx < NBOX * MPAD * MPAD; idx += NTHR) {
    unsigned u   = (unsigned)idx;
    unsigned n   = u / (MPAD * MPAD);
    unsigned rem = u - n * (MPAD * MPAD);
    unsigned x   = rem / MPAD;
    unsigned y   = rem - x * MPAD;
    float v = 0.0f;
    if (x > 0u && x < MPAD - 1 && y > 0u && y < MPAD - 1)
      v = (sraw[n * (MDIM * MDIM) + (x - 1) * MDIM + (y - 1)] + 1.0f) * 0.5f;
    smp[idx] = v;
  }
  __syncthreads();

  // ---- Phase 4: evaluate pixels, stream out with NT stores ----------------
  float* obase = out + ((size_t)b * HIMG + row0) * WIMG;
  #pragma unroll
  for (int k = 0; k < CHUNKS; ++k) {
    int c = tid + k * NTHR;        // float4 index within block
    int p = c * 4;
    int r = p >> 10;               // row within block (W = 1024)
    int j = p & (WIMG - 1);        // column of first pixel
    unsigned mrow = srowmask[r];
    v4f o;
    #pragma unroll
    for (int q = 0; q < 4; ++q) {
      int col = j + q;
      float val = -1.0f;           // background: 0*2 - 1
      unsigned mm = mrow;
      while (mm) {                 // highest box index wins (scan overwrite)
        int n = 31 - __builtin_clz(mm);
        if (col >= sb1[n] && col <= sb3[n]) {
          int   x0 = srx0[n * ROWS + r], x1 = srx1[n * ROWS + r];
          float tx = srtx[n * ROWS + r];
          float s  = ((float)(col - sb1[n]) + 0.5f) * sfy[n] - 0.5f;
          s = fminf(fmaxf(s, 0.0f), (float)(MPAD - 1));
          float f  = floorf(s);
          int   y0 = (int)f;
          int   y1 = min(y0 + 1, MPAD - 1);
          float ty = s - f;
          const float* mpp = &smp[n * (MPAD * MPAD)];
          float a  = mpp[x0 * MPAD + y0];
          float bb = mpp[x0 * MPAD + y1];
          float cc = mpp[x1 * MPAD + y0];
          float dd = mpp[x1 * MPAD + y1];
          float r0 = fmaf(tx, cc - a, a);
          float r1 = fmaf(tx, dd - bb, bb);
          float v  = fmaf(ty, r1 - r0, r0);
          val = fmaf(v, 2.0f, -1.0f);
          break;
        }
        mm &= ~(1u << n);
      }
      o[q] = val;
    }
    __builtin_nontemporal_store(o, (v4f*)(obase + r * WIMG + j));
  }
}

extern "C" void kernel_launch(void* const* d_in, const int* in_sizes, int n_in,
                              void* d_out, int out_size, void* d_ws, size_t ws_size,
                              hipStream_t stream) {
  (void)in_sizes; (void)n_in; (void)out_size; (void)d_ws; (void)ws_size;
  const float* masks = (const float*)d_in[0];   // [32,16,1,28,28] f32
  const float* rects = (const float*)d_in[1];   // [32,16,4] f32
  float*       out   = (float*)d_out;           // [32,1,1024,1024] f32
  dim3 grid(BIMG * (HIMG / ROWS));              // 1024 blocks
  dim3 block(NTHR);
  paste_masks_kernel<<<grid, block, 0, stream>>>(masks, rects, out);
}